// CopyModule_70360154243113
// MI455X (gfx1250) — compile-verified
//
#include <hip/hip_runtime.h>
#include <math.h>

#define Bsz 4
#define Hh  16
#define Tt  512
#define Ss  512
#define Dd  1024
#define Vv  32000
#define CHUNK 4096

typedef __attribute__((ext_vector_type(2))) float v2f;
typedef __attribute__((ext_vector_type(8))) float v8f;

// ---------------------------------------------------------------------------
// K1: c_att[b,t,s] = mean_h decoder_attention[b,h,t,s]
// ---------------------------------------------------------------------------
__global__ __launch_bounds__(256)
void k_mean_heads(const float* __restrict__ att, float* __restrict__ catt) {
    unsigned i = blockIdx.x * 256u + threadIdx.x;          // < B*T*S = 1M
    unsigned b = i / (Tt * Ss);
    unsigned r = i % (Tt * Ss);
    const float* p = att + (size_t)b * Hh * Tt * Ss + r;
    float s = 0.f;
#pragma unroll
    for (int h = 0; h < Hh; ++h) s += p[(size_t)h * Tt * Ss];
    catt[i] = s * (1.0f / Hh);
}

// ---------------------------------------------------------------------------
// K2: ctx[b,t,d] = sum_s c_att[b,t,s] * enc[b,s,d]   via V_WMMA_F32_16X16X4_F32
// One wave per 16x16 output tile; K-loop over S in steps of 4.
// A 16x4 f32: lane<16 -> M=lane, K={k,k+1}; lane>=16 -> M=lane-16, K={k+2,k+3}
// B 4x16 f32: symmetric (N = lane&15, K pair selected by lane>>4)
// C/D 16x16 f32: VGPR r -> M = r + 8*(lane>>4), N = lane&15
// ---------------------------------------------------------------------------
__global__ __launch_bounds__(32)
void k_ctx_wmma(const float* __restrict__ catt, const float* __restrict__ enc,
                float* __restrict__ ctx) {
    const int d0   = blockIdx.x * 16;
    const int t0   = blockIdx.y * 16;
    const int b    = blockIdx.z;
    const int lane = threadIdx.x;
    const int half = lane >> 4;
    const int lm   = lane & 15;

    const float* A  = catt + (size_t)b * Tt * Ss;
    const float* Bm = enc  + (size_t)b * Ss * Dd;

    v8f c = {};
    for (int k = 0; k < Ss; k += 4) {
        const int ka = k + 2 * half;
        v2f a, bb;
        a.x  = A[(size_t)(t0 + lm) * Ss + ka];
        a.y  = A[(size_t)(t0 + lm) * Ss + ka + 1];
        bb.x = Bm[(size_t)ka       * Dd + d0 + lm];
        bb.y = Bm[(size_t)(ka + 1) * Dd + d0 + lm];
        c = __builtin_amdgcn_wmma_f32_16x16x4_f32(
                /*neg_a=*/false, a, /*neg_b=*/false, bb,
                /*c_mod=*/(short)0, c, /*reuse_a=*/false, /*reuse_b=*/false);
    }

    float* C = ctx + (size_t)b * Tt * Dd;
#pragma unroll
    for (int r = 0; r < 8; ++r) {
        const int row = t0 + r + 8 * half;
        C[(size_t)row * Dd + d0 + lm] = c[r];
    }
}

// ---------------------------------------------------------------------------
// K3: p_copy[b,t] = sigmoid( dlh.w_logits + die.w_embeds + ctx.w_enc + biases )
// ---------------------------------------------------------------------------
__global__ __launch_bounds__(256)
void k_gate(const float* __restrict__ dlh, const float* __restrict__ die,
            const float* __restrict__ ctx,
            const float* __restrict__ wl, const float* __restrict__ bl,
            const float* __restrict__ we, const float* __restrict__ be,
            const float* __restrict__ wc, const float* __restrict__ bc,
            const float* __restrict__ bias, float* __restrict__ pcopy) {
    __shared__ float red[256];
    const int row = blockIdx.x;                 // b*T + t
    const int tid = threadIdx.x;
    const float* a = dlh + (size_t)row * Dd;
    const float* e = die + (size_t)row * Dd;
    const float* c = ctx + (size_t)row * Dd;
    float acc = 0.f;
    for (int d = tid; d < Dd; d += 256)
        acc += a[d] * wl[d] + e[d] * we[d] + c[d] * wc[d];
    red[tid] = acc;
    __syncthreads();
    for (int off = 128; off > 0; off >>= 1) {
        if (tid < off) red[tid] += red[tid + off];
        __syncthreads();
    }
    if (tid == 0) {
        const float g = red[0] + bl[0] + be[0] + bc[0] + bias[0];
        pcopy[row] = 1.0f / (1.0f + __expf(-g));
    }
}

// ---------------------------------------------------------------------------
// K4: out[b,t,v] = log( p*c_dist + (1-p)*softmax(logits) + 1e-12 )
// One workgroup per (b,t) row. Online softmax stats, then stream V in LDS
// chunks; c_dist built by fixed-point u64 LDS scatter-add (deterministic).
// ---------------------------------------------------------------------------
__global__ __launch_bounds__(256)
void k_output(const float* __restrict__ logits, const int* __restrict__ ids,
              const float* __restrict__ catt, const float* __restrict__ pcopy,
              float* __restrict__ out) {
    __shared__ unsigned long long cchunk[CHUNK];
    __shared__ float redm[256];
    __shared__ float reds[256];

    const int row = blockIdx.x;                 // b*T + t
    const int b   = row / Tt;
    const int tid = threadIdx.x;
    const float* lrow = logits + (size_t)row * Vv;
    float*       orow = out    + (size_t)row * Vv;

    // my two scatter pairs (S=512, 256 threads)
    const int   id0 = ids[b * Ss + tid];
    const int   id1 = ids[b * Ss + tid + 256];
    const float va0 = catt[(size_t)row * Ss + tid];
    const float va1 = catt[(size_t)row * Ss + tid + 256];
    const unsigned long long q0 = (unsigned long long)((double)va0 * 4294967296.0);
    const unsigned long long q1 = (unsigned long long)((double)va1 * 4294967296.0);

    // online softmax stats over this thread's strided slice
    float m = -INFINITY, s = 0.f;
    for (int v = tid; v < Vv; v += 256) {
        const float x = lrow[v];
        if (x > m) { s = s * __expf(m - x) + 1.0f; m = x; }
        else       { s += __expf(x - m); }
    }
    redm[tid] = m; reds[tid] = s;
    __syncthreads();
    for (int off = 128; off > 0; off >>= 1) {
        if (tid < off) {
            const float m1 = redm[tid], s1 = reds[tid];
            const float m2 = redm[tid + off], s2 = reds[tid + off];
            const float M = fmaxf(m1, m2);
            reds[tid] = s1 * __expf(m1 - M) + s2 * __expf(m2 - M);
            redm[tid] = M;
        }
        __syncthreads();
    }
    const float M    = redm[0];
    const float invS = 1.0f / reds[0];
    const float p    = pcopy[row];
    const float q    = 1.0f - p;
    const float deq  = 2.3283064365386963e-10f;   // 2^-32

    for (int v0 = 0; v0 < Vv; v0 += CHUNK) {
        for (int i = tid; i < CHUNK; i += 256) cchunk[i] = 0ull;
        __syncthreads();
        if (id0 >= v0 && id0 < v0 + CHUNK)
            atomicAdd(&cchunk[id0 - v0], q0);     // ds_add_u64, order-independent
        if (id1 >= v0 && id1 < v0 + CHUNK)
            atomicAdd(&cchunk[id1 - v0], q1);
        __syncthreads();
        const int vend = min(v0 + CHUNK, Vv);
        for (int v = v0 + tid; v < vend; v += 256) {
            const float g = __expf(lrow[v] - M) * invS;
            const float c = (float)cchunk[v - v0] * deq;
            orow[v] = logf(p * c + q * g + 1e-12f);
        }
        __syncthreads();
    }
}

// ---------------------------------------------------------------------------
extern "C" void kernel_launch(void* const* d_in, const int* in_sizes, int n_in,
                              void* d_out, int out_size, void* d_ws, size_t ws_size,
                              hipStream_t stream) {
    const float* dec_att = (const float*)d_in[0];
    const float* dlh     = (const float*)d_in[1];
    const int*   ids     = (const int*)  d_in[2];
    const float* logits  = (const float*)d_in[3];
    const float* die     = (const float*)d_in[4];
    const float* enc     = (const float*)d_in[5];
    const float* w_l     = (const float*)d_in[6];
    const float* b_l     = (const float*)d_in[7];
    const float* w_e     = (const float*)d_in[8];
    const float* b_e     = (const float*)d_in[9];
    const float* w_c     = (const float*)d_in[10];
    const float* b_c     = (const float*)d_in[11];
    const float* bias    = (const float*)d_in[12];

    float* ws   = (float*)d_ws;
    float* catt = ws;                                   // B*T*S floats (4 MB)
    float* ctx  = catt + (size_t)Bsz * Tt * Ss;         // B*T*D floats (8 MB)
    float* pcp  = ctx  + (size_t)Bsz * Tt * Dd;         // B*T floats

    k_mean_heads<<<(Bsz * Tt * Ss) / 256, 256, 0, stream>>>(dec_att, catt);

    dim3 g2(Dd / 16, Tt / 16, Bsz);
    k_ctx_wmma<<<g2, 32, 0, stream>>>(catt, enc, ctx);

    k_gate<<<Bsz * Tt, 256, 0, stream>>>(dlh, die, ctx,
                                         w_l, b_l, w_e, b_e, w_c, b_c, bias, pcp);

    k_output<<<Bsz * Tt, 256, 0, stream>>>(logits, ids, catt, pcp, (float*)d_out);
}